// GraphResBlock_86346022519488
// MI455X (gfx1250) — compile-verified
//
#include <hip/hip_runtime.h>
#include <hip/hip_bf16.h>

typedef __attribute__((ext_vector_type(2))) float v2f;
typedef __attribute__((ext_vector_type(8))) float v8f;

#define C_CH 256
#define NET 7
#define KDIM (NET * C_CH)   // 1792
#define EPS_GN 1e-5f

// ---------------------------------------------------------------------------
// GroupNorm stats: per-block partial sums over 64 sorted-by-batch rows,
// flushed with atomics into S1/S2 [8][256] and counts[8].
// ---------------------------------------------------------------------------
__global__ void __launch_bounds__(256) gn_stats_kernel(
    const float* __restrict__ x, const int* __restrict__ bid,
    float* __restrict__ S1, float* __restrict__ S2, float* __restrict__ counts,
    int N) {
  __shared__ int bids[64];
  int n0 = blockIdx.x * 64;
  int rows = N - n0; if (rows > 64) rows = 64;
  int c = threadIdx.x;  // channel 0..255
  if (c < rows) bids[c] = bid[n0 + c];
  __syncthreads();
  float s1 = 0.f, s2 = 0.f;
  int cur = bids[0];
  int segstart = 0;
  for (int r = 0; r < rows; ++r) {
    int b = bids[r];
    if (b != cur) {
      atomicAdd(&S1[cur * C_CH + c], s1);
      atomicAdd(&S2[cur * C_CH + c], s2);
      if (c == 0) atomicAdd(&counts[cur], (float)(r - segstart));
      s1 = 0.f; s2 = 0.f; cur = b; segstart = r;
    }
    float v = x[(long)(n0 + r) * C_CH + c];
    s1 += v; s2 += v * v;
  }
  atomicAdd(&S1[cur * C_CH + c], s1);
  atomicAdd(&S2[cur * C_CH + c], s2);
  if (c == 0) atomicAdd(&counts[cur], (float)(rows - segstart));
}

// One block, 256 threads = 8 batches x 32 groups.
__global__ void gn_finalize_kernel(const float* __restrict__ S1,
                                   const float* __restrict__ S2,
                                   const float* __restrict__ counts,
                                   float* __restrict__ mean,
                                   float* __restrict__ invstd) {
  int t = threadIdx.x;
  int b = t >> 5, g = t & 31;
  float nb = counts[b];
  float cntf = nb * 8.0f;               // count * cpg
  float invc = 1.0f / (cntf + EPS_GN);
  float s1 = 0.f, s2 = 0.f;
  for (int j = 0; j < 8; ++j) {
    s1 += S1[b * C_CH + g * 8 + j];
    s2 += S2[b * C_CH + g * 8 + j];
  }
  float m = s1 * invc;
  // sum over group of (x-m)^2 = s2 - 2*m*s1 + cpg*n_b*m^2
  float var = (s2 - 2.0f * m * s1 + 8.0f * nb * m * m) * invc;
  mean[t] = m;
  invstd[t] = rsqrtf(var + EPS_GN);
}

// Fused (x - mean)*invstd*w + b, then SiLU. float4 per thread.
__global__ void __launch_bounds__(256) gn_apply_silu_kernel(
    const float* __restrict__ x, const int* __restrict__ bid,
    const float* __restrict__ mean, const float* __restrict__ invstd,
    const float* __restrict__ w, const float* __restrict__ bb,
    float* __restrict__ out, int N) {
  long i = (long)blockIdx.x * 256 + threadIdx.x;   // one float4
  long total = (long)N * (C_CH / 4);
  if (i >= total) return;
  int n = (int)(i >> 6);
  int c4 = (int)(i & 63);
  int b = bid[n];
  int g = (c4 * 4) >> 3;   // 4-aligned chunk stays inside one 8-wide group
  float m = mean[b * 32 + g];
  float is = invstd[b * 32 + g];
  float4 xv = ((const float4*)x)[i];
  float4 wv = ((const float4*)w)[c4];
  float4 bv = ((const float4*)bb)[c4];
  float4 o;
  float y;
  y = (xv.x - m) * is * wv.x + bv.x; o.x = y / (1.0f + __expf(-y));
  y = (xv.y - m) * is * wv.y + bv.y; o.y = y / (1.0f + __expf(-y));
  y = (xv.z - m) * is * wv.z + bv.z; o.z = y / (1.0f + __expf(-y));
  y = (xv.w - m) * is * wv.w + bv.w; o.w = y / (1.0f + __expf(-y));
  ((float4*)out)[i] = o;
}

// ---------------------------------------------------------------------------
// Edge scatter: 64 threads per edge; gather h[col] (L2-resident) and
// atomicAdd into col_data[(row*7+etype)*256 + c]. Lane 0 bumps the count.
// ---------------------------------------------------------------------------
__global__ void __launch_bounds__(256) scatter_kernel(
    const float* __restrict__ h, const int* __restrict__ row,
    const int* __restrict__ col, const int* __restrict__ et,
    float* __restrict__ cdat, float* __restrict__ cnt, int E) {
  long t = (long)blockIdx.x * 256 + threadIdx.x;
  long total = (long)E * 64;
  if (t >= total) return;
  int e = (int)(t >> 6);
  int c4 = (int)(t & 63);
  int r = row[e], cc = col[e], ty = et[e];
  float4 v = ((const float4*)h)[(long)cc * 64 + c4];
  long base = ((long)r * NET + ty) * C_CH + c4 * 4;
  atomicAdd(&cdat[base + 0], v.x);
  atomicAdd(&cdat[base + 1], v.y);
  atomicAdd(&cdat[base + 2], v.z);
  atomicAdd(&cdat[base + 3], v.w);
  if (c4 == 0) atomicAdd(&cnt[(long)r * NET + ty], 1.0f);
}

__global__ void invcnt_kernel(const float* __restrict__ cnt,
                              float* __restrict__ invc, int n) {
  int i = blockIdx.x * 256 + threadIdx.x;
  if (i < n) invc[i] = 1.0f / fmaxf(cnt[i], 1.0f);
}

// ---------------------------------------------------------------------------
// WMMA GEMM: out[N,256] = (col_data[N,1792] scaled by invc) @ W[1792,256]
// Block: 256 threads (8 waves), BM=32 rows, BN=256 cols; wave w owns 32 cols.
// A chunk (32x64) staged in LDS (scatter-mean 1/cnt folded into the load);
// K stepped by 4 with v_wmma_f32_16x16x4_f32; B frags read from L2-resident W.
// Optional residual (+x) fused into the epilogue.
// ---------------------------------------------------------------------------
__global__ void __launch_bounds__(256) gemm_wmma_kernel(
    const float* __restrict__ A,      // col_data, [N*1792]
    const float* __restrict__ invc,   // [N*7]
    const float* __restrict__ W,      // [1792*256]
    const float* __restrict__ resid,  // nullable, [N*256]
    float* __restrict__ out, int N) {
  __shared__ float Alds[32][68];      // +4 pad: lanes m=0..15 hit distinct banks
  int m0 = blockIdx.x * 32;
  int tid = threadIdx.x;
  int wv = tid >> 5;                  // wave 0..7
  int lane = tid & 31;
  int mloc = lane & 15;               // A/B fragment: M (or N) = lane%16
  int kp = (lane >> 4) << 1;          // lane half selects K pair {0,1} or {2,3}
  int ncol = wv * 32 + (lane & 15);   // this wave's first N-tile column

  v8f acc00{}, acc01{}, acc10{}, acc11{};

  for (int kb = 0; kb < KDIM; kb += 64) {
    int etyp = kb >> 8;               // 64-wide window never crosses a 256 bound
    __syncthreads();
    for (int i = tid; i < 32 * 64; i += 256) {
      int r = i >> 6, cc = i & 63;
      int rr = m0 + r; if (rr >= N) rr = N - 1;
      float s = invc[(long)rr * NET + etyp];
      Alds[r][cc] = A[(long)rr * KDIM + kb + cc] * s;
    }
    __syncthreads();
    for (int kk = 0; kk < 64; kk += 4) {
      v2f a0, a1, b0, b1;
      a0[0] = Alds[mloc][kk + kp];
      a0[1] = Alds[mloc][kk + kp + 1];
      a1[0] = Alds[mloc + 16][kk + kp];
      a1[1] = Alds[mloc + 16][kk + kp + 1];
      long krow = (long)(kb + kk + kp);
      b0[0] = W[krow * C_CH + ncol];
      b0[1] = W[(krow + 1) * C_CH + ncol];
      b1[0] = W[krow * C_CH + ncol + 16];
      b1[1] = W[(krow + 1) * C_CH + ncol + 16];
      acc00 = __builtin_amdgcn_wmma_f32_16x16x4_f32(false, a0, false, b0, (short)0, acc00, false, false);
      acc01 = __builtin_amdgcn_wmma_f32_16x16x4_f32(false, a0, false, b1, (short)0, acc01, false, false);
      acc10 = __builtin_amdgcn_wmma_f32_16x16x4_f32(false, a1, false, b0, (short)0, acc10, false, false);
      acc11 = __builtin_amdgcn_wmma_f32_16x16x4_f32(false, a1, false, b1, (short)0, acc11, false, false);
    }
  }

  // D layout: VGPR j -> row (lane>>4)*8 + j, col = lane&15 (per 16x16 tile)
  int ro = (lane >> 4) * 8;
  int colg = wv * 32 + (lane & 15);
  for (int j = 0; j < 8; ++j) {
    int r0 = m0 + ro + j;
    int r1 = m0 + 16 + ro + j;
    float v00 = acc00[j], v01 = acc01[j], v10 = acc10[j], v11 = acc11[j];
    if (resid) {
      if (r0 < N) { v00 += resid[(long)r0 * C_CH + colg];
                    v01 += resid[(long)r0 * C_CH + colg + 16]; }
      if (r1 < N) { v10 += resid[(long)r1 * C_CH + colg];
                    v11 += resid[(long)r1 * C_CH + colg + 16]; }
    }
    if (r0 < N) {
      out[(long)r0 * C_CH + colg] = v00;
      out[(long)r0 * C_CH + colg + 16] = v01;
    }
    if (r1 < N) {
      out[(long)r1 * C_CH + colg] = v10;
      out[(long)r1 * C_CH + colg + 16] = v11;
    }
  }
}

// ---------------------------------------------------------------------------
extern "C" void kernel_launch(void* const* d_in, const int* in_sizes, int n_in,
                              void* d_out, int out_size, void* d_ws, size_t ws_size,
                              hipStream_t stream) {
  const float* x       = (const float*)d_in[0];
  const float* gn1_w   = (const float*)d_in[1];
  const float* gn1_b   = (const float*)d_in[2];
  const float* conv1_w = (const float*)d_in[3];
  const float* gn2_w   = (const float*)d_in[4];
  const float* gn2_b   = (const float*)d_in[5];
  const float* conv2_w = (const float*)d_in[6];
  const int* edge_index = (const int*)d_in[7];
  const int* edge_type  = (const int*)d_in[8];
  const int* batch_id   = (const int*)d_in[9];

  const int N = in_sizes[0] / C_CH;
  const int E = in_sizes[7] / 2;
  const int N7 = N * NET;
  const int* row = edge_index;
  const int* col = edge_index + E;

  char* ws = (char*)d_ws;
  size_t off = 0;
  auto carve = [&](size_t sz) -> char* {
    char* p = ws + off;
    off = (off + sz + 255) & ~(size_t)255;
    return p;
  };
  float* S1     = (float*)carve(8 * C_CH * sizeof(float));
  float* S2     = (float*)carve(8 * C_CH * sizeof(float));
  float* counts = (float*)carve(8 * sizeof(float));
  size_t statsBytes = off;                       // S1,S2,counts contiguous at ws[0]
  float* mean   = (float*)carve(8 * 32 * sizeof(float));
  float* invstd = (float*)carve(8 * 32 * sizeof(float));
  float* cnt    = (float*)carve((size_t)N7 * sizeof(float));
  float* invc   = (float*)carve((size_t)N7 * sizeof(float));
  float* h      = (float*)carve((size_t)N * C_CH * sizeof(float));
  float* t      = (float*)carve((size_t)N * C_CH * sizeof(float));
  float* cdat   = (float*)carve((size_t)N7 * C_CH * sizeof(float));
  (void)mean; (void)invstd;

  int gnBlocks = (N + 63) / 64;
  long applyTotal = (long)N * (C_CH / 4);
  int applyBlocks = (int)((applyTotal + 255) / 256);
  long scatTotal = (long)E * 64;
  int scatBlocks = (int)((scatTotal + 255) / 256);
  int invBlocks = (N7 + 255) / 256;
  int gemmBlocks = (N + 31) / 32;
  size_t cdatBytes = (size_t)N7 * C_CH * sizeof(float);

  // ---------------- layer 1 ----------------
  hipMemsetAsync(ws, 0, statsBytes, stream);
  gn_stats_kernel<<<gnBlocks, 256, 0, stream>>>(x, batch_id, S1, S2, counts, N);
  gn_finalize_kernel<<<1, 256, 0, stream>>>(S1, S2, counts, mean, invstd);
  gn_apply_silu_kernel<<<applyBlocks, 256, 0, stream>>>(x, batch_id, mean, invstd,
                                                        gn1_w, gn1_b, h, N);
  hipMemsetAsync(cdat, 0, cdatBytes, stream);
  hipMemsetAsync(cnt, 0, (size_t)N7 * sizeof(float), stream);
  scatter_kernel<<<scatBlocks, 256, 0, stream>>>(h, row, col, edge_type, cdat, cnt, E);
  invcnt_kernel<<<invBlocks, 256, 0, stream>>>(cnt, invc, N7);
  gemm_wmma_kernel<<<gemmBlocks, 256, 0, stream>>>(cdat, invc, conv1_w, nullptr, t, N);

  // ---------------- layer 2 ----------------
  hipMemsetAsync(ws, 0, statsBytes, stream);
  gn_stats_kernel<<<gnBlocks, 256, 0, stream>>>(t, batch_id, S1, S2, counts, N);
  gn_finalize_kernel<<<1, 256, 0, stream>>>(S1, S2, counts, mean, invstd);
  gn_apply_silu_kernel<<<applyBlocks, 256, 0, stream>>>(t, batch_id, mean, invstd,
                                                        gn2_w, gn2_b, h, N);
  hipMemsetAsync(cdat, 0, cdatBytes, stream);
  hipMemsetAsync(cnt, 0, (size_t)N7 * sizeof(float), stream);
  scatter_kernel<<<scatBlocks, 256, 0, stream>>>(h, row, col, edge_type, cdat, cnt, E);
  invcnt_kernel<<<invBlocks, 256, 0, stream>>>(cnt, invc, N7);
  gemm_wmma_kernel<<<gemmBlocks, 256, 0, stream>>>(cdat, invc, conv2_w, x,
                                                   (float*)d_out, N);
}